// LPMaskLoss_17265768529972
// MI455X (gfx1250) — compile-verified
//
#include <hip/hip_runtime.h>

typedef __attribute__((ext_vector_type(4))) float f4;
typedef __attribute__((ext_vector_type(2))) float v2f;
typedef __attribute__((ext_vector_type(8))) float v8f;

#define HW_     65536     // 256*256
#define B_      32
#define G_      2
#define CHUNKS  16        // chunks per (b,g) plane
#define TPB     256       // 8 waves of 32

// ---------------------------------------------------------------------------
// Stage 1: streaming pass. grid = B*G*CHUNKS blocks; block (plane,chunk) owns
// 4096 pixels of plane (b,g). Each thread: 4 iterations x float4 (16 pixels),
// 8 x B128 non-temporal loads per iteration (4 out channels + 4 tar channels).
// Reduces {msum, count, dsum, bce} to one 4-float partial per block (fixed
// slot in scratch -> fully deterministic, no atomics).
// ---------------------------------------------------------------------------
__global__ __launch_bounds__(TPB) void lpmask_stage1(
    const float* __restrict__ outp, const float* __restrict__ targ,
    float* __restrict__ part) {
  const int chunk = blockIdx.x & (CHUNKS - 1);
  const int plane = blockIdx.x >> 4;          // plane = b*2 + g
  const int g = plane & 1;
  const int b = plane >> 1;
  const size_t base = (size_t)(b * 8 + 4 * g) * (size_t)HW_;
  const float* o0 = outp + base;
  const float* t0 = targ + base;

  float msum = 0.0f, cnt = 0.0f, dsum = 0.0f, bce = 0.0f;

  const int p0 = chunk * 4096 + (int)threadIdx.x * 4;
#pragma unroll
  for (int it = 0; it < 4; ++it) {
    const int p = p0 + it * 1024;
    f4 ox = __builtin_nontemporal_load((const f4*)(o0 + 0 * HW_ + p));
    f4 oy = __builtin_nontemporal_load((const f4*)(o0 + 1 * HW_ + p));
    f4 oz = __builtin_nontemporal_load((const f4*)(o0 + 2 * HW_ + p));
    f4 om = __builtin_nontemporal_load((const f4*)(o0 + 3 * HW_ + p));
    f4 tx = __builtin_nontemporal_load((const f4*)(t0 + 0 * HW_ + p));
    f4 ty = __builtin_nontemporal_load((const f4*)(t0 + 1 * HW_ + p));
    f4 tz = __builtin_nontemporal_load((const f4*)(t0 + 2 * HW_ + p));
    f4 tm = __builtin_nontemporal_load((const f4*)(t0 + 3 * HW_ + p));
#pragma unroll
    for (int j = 0; j < 4; ++j) {
      // BCE on sigmoid(mask), logs clipped at -100 like the reference
      const float x  = om[j];
      const float pm = 1.0f / (1.0f + __expf(-x));
      float lp  = fmaxf(__logf(pm),  -100.0f);
      float l1  = fmaxf(log1pf(-pm), -100.0f);
      const float t = tm[j];
      bce -= t * lp + (1.0f - t) * l1;
      // L2 norm of image-channel diff
      const float d0 = ox[j] - tx[j];
      const float d1 = oy[j] - ty[j];
      const float d2 = oz[j] - tz[j];
      const float dn = __fsqrt_rn(__builtin_fmaf(d0, d0, __builtin_fmaf(d1, d1, d2 * d2)));
      dsum += dn;
      if (pm > 0.7f) {
        msum += dn;
        cnt  += (dn != 0.0f) ? 1.0f : 0.0f;   // masked != 0 semantics
      }
    }
  }

  // wave32 reduction
#pragma unroll
  for (int off = 16; off > 0; off >>= 1) {
    msum += __shfl_xor(msum, off, 32);
    cnt  += __shfl_xor(cnt,  off, 32);
    dsum += __shfl_xor(dsum, off, 32);
    bce  += __shfl_xor(bce,  off, 32);
  }

  __shared__ float red[8][4];
  const int lane = (int)threadIdx.x & 31;
  const int wv   = (int)threadIdx.x >> 5;
  if (lane == 0) {
    red[wv][0] = msum; red[wv][1] = cnt; red[wv][2] = dsum; red[wv][3] = bce;
  }
  __syncthreads();
  if (threadIdx.x == 0) {
    float a = 0.f, c2 = 0.f, d2 = 0.f, e2 = 0.f;
#pragma unroll
    for (int w = 0; w < 8; ++w) {
      a  += red[w][0]; c2 += red[w][1]; d2 += red[w][2]; e2 += red[w][3];
    }
    float* dst = part + (size_t)blockIdx.x * 4;   // slot = plane*16 + chunk
    dst[0] = a; dst[1] = c2; dst[2] = d2; dst[3] = e2;
  }
}

// ---------------------------------------------------------------------------
// Stage 2: one wave32. Each lane deterministically folds the 16 chunk
// partials of two (b,g) entries, computes per_sample and bce totals, then two
// V_WMMA_F32_16X16X4_F32 (A = values, B = ones) reduce all 32 batch entries
// per group: rows 0-7 of A hold group 0 (b = 4*row + k), rows 8-15 group 1.
// Row sums land in the C/D layout: lanes 0-15 hold rows 0-7 (group 0) in
// acc[0..7], lanes 16-31 hold rows 8-15 (group 1).
// ---------------------------------------------------------------------------
__global__ __launch_bounds__(32) void lpmask_stage2(
    const float* __restrict__ part, float* __restrict__ out) {
  const int L  = (int)threadIdx.x;       // 0..31, EXEC all ones (WMMA req.)
  const int m  = L & 15;                 // A-matrix row
  const int kb = (L >> 4) << 1;          // lanes 0-15 hold k=0,1; 16-31 k=2,3
  const int g  = m >> 3;
  const int r  = m & 7;

  v2f a_ps, a_bce;
#pragma unroll
  for (int e = 0; e < 2; ++e) {
    const int b     = 4 * r + kb + e;    // A[m][kb+e] = value(b, g)
    const int plane = b * 2 + g;
    float msum = 0.f, cnt = 0.f, dsum = 0.f, bce = 0.f;
#pragma unroll
    for (int c = 0; c < CHUNKS; ++c) {
      const float* s = part + (size_t)(plane * CHUNKS + c) * 4;
      msum += s[0]; cnt += s[1]; dsum += s[2]; bce += s[3];
    }
    const float fallback = dsum * (1.0f / (float)HW_);
    const float ps = (cnt > 0.0f) ? (msum / fmaxf(cnt, 1.0f)) : fallback;
    a_ps[e]  = ps;
    a_bce[e] = bce;
  }

  v2f ones; ones[0] = 1.0f; ones[1] = 1.0f;
  v8f zero = {};
  // D = A x ones + 0 : each row of D = sum_k A[m][k], replicated over columns
  v8f dps = __builtin_amdgcn_wmma_f32_16x16x4_f32(
      false, a_ps,  false, ones, (short)0, zero, false, false);
  v8f dbc = __builtin_amdgcn_wmma_f32_16x16x4_f32(
      false, a_bce, false, ones, (short)0, zero, false, false);

  float sps = 0.f, sbc = 0.f;
#pragma unroll
  for (int i = 0; i < 8; ++i) { sps += dps[i]; sbc += dbc[i]; }
  // lanes 0-15: group-0 totals over all b; lanes 16-31: group-1 totals

  const float mask_loss = sbc * (1.0f / ((float)B_ * (float)HW_));
  const float nocs_loss = sps * (1.0f / (float)B_);
  const float partl = 0.7f * mask_loss + 0.3f * nocs_loss;

  const float other = __shfl(partl, 16, 32);
  if (L == 0) out[0] = 0.5f * (partl + other);
}

// ---------------------------------------------------------------------------
extern "C" void kernel_launch(void* const* d_in, const int* in_sizes, int n_in,
                              void* d_out, int out_size, void* d_ws, size_t ws_size,
                              hipStream_t stream) {
  const float* outp = (const float*)d_in[0];   // 'output' (B,C,H,W) f32
  const float* targ = (const float*)d_in[1];   // 'target' (B,C,H,W) f32
  float* part = (float*)d_ws;                  // 1024 blocks * 4 floats = 16 KB

  lpmask_stage1<<<B_ * G_ * CHUNKS, TPB, 0, stream>>>(outp, targ, part);
  lpmask_stage2<<<1, 32, 0, stream>>>(part, (float*)d_out);
}